// Model5_Alation_54185307406496
// MI455X (gfx1250) — compile-verified
//
#include <hip/hip_runtime.h>
#include <math.h>

// ---------------- problem constants ----------------
#define BB   16
#define NN   325
#define T0   192
#define T1   96
#define T2   48
#define FIN  3
#define DM   64
#define TOUT 12
#define FR0  97       // 192/2+1
#define FR1  49
#define FR2  25
#define MP0  208      // fwd-basis rows: 2*FR padded up to mult of 16
#define MP1  112
#define MP2  64
#define KP0  196      // inv-basis cols: 2*FR padded up to mult of 4
#define KP1  100
#define KP2  52
#define LSTR 80       // LDS row stride (floats); 2*80 % 64 == 32 -> conflict-free B frags
#define ROWS_E    336 // 192+96+48
#define ROWS_SPEC 208

typedef float v2f __attribute__((ext_vector_type(2)));
typedef float v8f __attribute__((ext_vector_type(8)));

// ---------------------------------------------------------------------------
// D(M x 64, LDS stride LSTR) = A(M x K, global row-major lda) * B(K x 64, LDS)
// epilogue: optional per-row bias, exact GELU, residual add from LDS.
// Fragment layout per CDNA5 ISA 7.12.2 (wave32, f32 16x16x4):
//   A: lane = M row (lane&15), K = 2*(lane>>4) + vgpr
//   B: lane = N col (lane&15), K = 2*(lane>>4) + vgpr
//   C/D: col = lane&15, vgpr v -> row v + 8*(lane>>4)
// ---------------------------------------------------------------------------
__device__ __forceinline__ void mm_wmma(
    const float* __restrict__ A, int lda, int M, int K,
    const float* __restrict__ Bl, float* __restrict__ Dl,
    const float* __restrict__ bias, int do_gelu,
    const float* __restrict__ Rl)
{
  const int lane = threadIdx.x & 31;
  const int wave = threadIdx.x >> 5;
  const int r    = lane & 15;
  const int half = lane >> 4;
  const int ntiles = (M >> 4) << 2;          // (M/16) row tiles x 4 col tiles
  for (int tile = wave; tile < ntiles; tile += 8) {
    const int tm = (tile >> 2) << 4;
    const int tn = (tile & 3) << 4;
    v8f acc = {0.f,0.f,0.f,0.f,0.f,0.f,0.f,0.f};
    const float* Ar = A  + (size_t)(tm + r) * lda + 2*half;
    const float* Bc = Bl + tn + r + (size_t)(2*half) * LSTR;
    for (int k = 0; k < K; k += 4) {
      v2f a; a.x = Ar[k];               a.y = Ar[k + 1];
      v2f b; b.x = Bc[(size_t)k * LSTR]; b.y = Bc[(size_t)k * LSTR + LSTR];
      acc = __builtin_amdgcn_wmma_f32_16x16x4_f32(
              false, a, false, b, (short)0, acc, false, false);
    }
    float* Dp = Dl + tn + r;
    #pragma unroll
    for (int v = 0; v < 8; ++v) {
      const int row = tm + v + 8*half;
      float val = acc[v];
      if (bias)    val += bias[row];
      if (do_gelu) val  = 0.5f * val * (1.0f + erff(val * 0.7071067811865475f));
      if (Rl)      val += Rl[(size_t)row * LSTR + tn + r];
      Dp[(size_t)row * LSTR] = val;
    }
  }
}

// ---------------- DFT basis generation (into d_ws, cached in L2) -----------
__global__ void gen_fwd_basis(float* dst, int FR, int Mpad, int T) {
  int idx = blockIdx.x * blockDim.x + threadIdx.x;
  if (idx >= Mpad * T) return;
  int m = idx / T, t = idx - m * T;
  float v = 0.f;
  if (m < 2 * FR) {
    int f  = (m < FR) ? m : (m - FR);
    int ph = (f * t) % T;                          // exact phase reduction
    float th = 6.283185307179586f * (float)ph / (float)T;
    v = (m < FR) ? cosf(th) : -sinf(th);           // X[f] = sum x[t] e^{-i th}
  }
  dst[idx] = v;
}

__global__ void gen_inv_basis(float* dst, int FR, int Kpad, int T) {
  int idx = blockIdx.x * blockDim.x + threadIdx.x;
  if (idx >= T * Kpad) return;
  int t = idx / Kpad, c = idx - t * Kpad;
  float v = 0.f;
  if (c < 2 * FR) {
    int f = (c < FR) ? c : (c - FR);
    float w = (f == 0 || 2 * f == T) ? 1.f : 2.f;  // rfft hermitian weights
    int ph = (f * t) % T;
    float th = 6.283185307179586f * (float)ph / (float)T;
    v = ((c < FR) ? w * cosf(th) : -w * sinf(th)) / (float)T;
  }
  dst[idx] = v;
}

// ---------------- main fused model kernel: 1 workgroup per (b,n) -----------
struct Params {
  const float* x;
  const float* w_emb;
  const float* b_emb;
  const float* fw1[2][2]; const float* fb1[2][2];
  const float* fw2[2][2]; const float* fb2[2][2];
  const float* c1w[3]; const float* c1b[3];
  const float* c2w[3]; const float* c2b[3];
  const float* fwdB[3];
  const float* invB[3];
  float* out;
};

__global__ void __launch_bounds__(256)
timemixer_kernel(Params p)
{
  extern __shared__ float smem[];
  float* ebuf  = smem;                         // x / trend / next-layer x (336 rows)
  float* sbuf  = ebuf + ROWS_E * LSTR;         // seasons / cascade outputs (336 rows)
  float* spec  = sbuf + ROWS_E * LSTR;         // spectrum / hidden scratch (208 rows)
  float* thr2  = spec + ROWS_SPEC * LSTR;      // 64 squared thresholds
  float* out12 = thr2 + 64;                    // 12 output accumulators

  const int bn  = blockIdx.x;
  const int tid = threadIdx.x;

  const int Ts[3]   = {T0, T1, T2};
  const int FRs[3]  = {FR0, FR1, FR2};
  const int MPs[3]  = {MP0, MP1, MP2};
  const int KPs[3]  = {KP0, KP1, KP2};
  const int offs[3] = {0, T0, T0 + T1};

  // ---- stage raw series, embed, build pyramid (mean-then-linear == linear-then-mean)
  const float* xin = p.x + (size_t)bn * T0 * FIN;
  for (int i = tid; i < T0 * FIN; i += 256) spec[i] = xin[i];
  __syncthreads();
  for (int i = tid; i < T0 * DM; i += 256) {
    int t = i >> 6, d = i & 63;
    float v = p.b_emb[d];
    v += spec[t*FIN + 0] * p.w_emb[0*DM + d];
    v += spec[t*FIN + 1] * p.w_emb[1*DM + d];
    v += spec[t*FIN + 2] * p.w_emb[2*DM + d];
    ebuf[t * LSTR + d] = v;
  }
  __syncthreads();
  for (int i = tid; i < T1 * DM; i += 256) {
    int t = i >> 6, d = i & 63;
    ebuf[(T0 + t) * LSTR + d] = 0.5f * (ebuf[(2*t)*LSTR + d] + ebuf[(2*t+1)*LSTR + d]);
  }
  __syncthreads();
  for (int i = tid; i < T2 * DM; i += 256) {
    int t = i >> 6, d = i & 63;
    ebuf[(T0+T1+t)*LSTR + d] = 0.5f * (ebuf[(T0+2*t)*LSTR + d] + ebuf[(T0+2*t+1)*LSTR + d]);
  }
  __syncthreads();

  // ---- 2 PDM fusion layers
  for (int layer = 0; layer < 2; ++layer) {
    #pragma unroll
    for (int s = 0; s < 3; ++s) {
      const int T = Ts[s], FR = FRs[s];
      float* es = ebuf + offs[s] * LSTR;
      float* ss = sbuf + offs[s] * LSTR;

      // rfft as matmul: spec rows [0,FR)=Re, [FR,2FR)=Im, rest 0 (zero basis rows)
      mm_wmma(p.fwdB[s], T, MPs[s], T, es, spec, nullptr, 0, nullptr);
      __syncthreads();

      // per-channel 5th-largest squared amplitude (sqrt is monotone -> same mask)
      if (tid < DM) {
        const int d = tid;
        float m0=-1.f, m1=-1.f, m2=-1.f, m3=-1.f, m4=-1.f;
        for (int f = 0; f < FR; ++f) {
          float a = 0.f;
          if (f != 0) {
            float re = spec[f*LSTR + d], im = spec[(FR+f)*LSTR + d];
            a = re*re + im*im;
          }
          if (a > m4) {
            if      (a > m0) { m4=m3; m3=m2; m2=m1; m1=m0; m0=a; }
            else if (a > m1) { m4=m3; m3=m2; m2=m1; m1=a; }
            else if (a > m2) { m4=m3; m3=m2; m2=a; }
            else if (a > m3) { m4=m3; m3=a; }
            else             { m4=a; }
          }
        }
        thr2[d] = m4;
      }
      __syncthreads();

      // mask: where(amp < thr) -> 0  (DC amplitude forced 0, as in reference)
      for (int i = tid; i < FR * DM; i += 256) {
        int f = i >> 6, d = i & 63;
        float re = spec[f*LSTR + d], im = spec[(FR+f)*LSTR + d];
        float a = (f == 0) ? 0.f : (re*re + im*im);
        if (a < thr2[d]) { spec[f*LSTR + d] = 0.f; spec[(FR+f)*LSTR + d] = 0.f; }
      }
      __syncthreads();

      // irfft as matmul -> season
      mm_wmma(p.invB[s], KPs[s], T, KPs[s], spec, ss, nullptr, 0, nullptr);
      __syncthreads();

      // trend = x - season (in place)
      for (int i = tid; i < T * DM; i += 256) {
        int t = i >> 6, d = i & 63;
        es[t*LSTR + d] -= ss[t*LSTR + d];
      }
      __syncthreads();
    }

    // cross-scale cascade: seasons[i] = W2 @ gelu(W1 @ prev + b1) + b2 + seasons[i]
    mm_wmma(p.fw1[layer][0], T0, T1, T0, sbuf, spec, p.fb1[layer][0], 1, nullptr);
    __syncthreads();
    mm_wmma(p.fw2[layer][0], T1, T1, T1, spec, sbuf + T0*LSTR,
            p.fb2[layer][0], 0, sbuf + T0*LSTR);
    __syncthreads();
    mm_wmma(p.fw1[layer][1], T1, T2, T1, sbuf + T0*LSTR, spec, p.fb1[layer][1], 1, nullptr);
    __syncthreads();
    mm_wmma(p.fw2[layer][1], T2, T2, T2, spec, sbuf + (T0+T1)*LSTR,
            p.fb2[layer][1], 0, sbuf + (T0+T1)*LSTR);
    __syncthreads();

    // x = season_out + trend
    for (int i = tid; i < ROWS_E * DM; i += 256) {
      int t = i >> 6, d = i & 63;
      ebuf[t*LSTR + d] += sbuf[t*LSTR + d];
    }
    __syncthreads();
  }

  // ---- prediction heads (tiny; deterministic LDS reduction, no float atomics)
  if (tid < TOUT) out12[tid] = p.c2b[0][0] + p.c2b[1][0] + p.c2b[2][0];
  __syncthreads();
  #pragma unroll
  for (int s = 0; s < 3; ++s) {
    const int T = Ts[s];
    const float* es = ebuf + offs[s] * LSTR;
    for (int i = tid; i < TOUT * DM; i += 256) {
      int o = i >> 6, d = i & 63;
      const float* w = p.c1w[s] + o * T;
      float acc = 0.f;
      for (int t = 0; t < T; ++t) acc += es[t*LSTR + d] * w[t];
      spec[o*LSTR + d] = (acc + p.c1b[s][o]) * p.c2w[s][d];
    }
    __syncthreads();
    if (tid < TOUT) {
      float acc = 0.f;
      for (int d = 0; d < DM; ++d) acc += spec[tid*LSTR + d];
      out12[tid] += acc;
    }
    __syncthreads();
  }
  if (tid < TOUT) p.out[(size_t)bn * TOUT + tid] = out12[tid];
}

// ---------------------------------------------------------------------------
extern "C" void kernel_launch(void* const* d_in, const int* in_sizes, int n_in,
                              void* d_out, int out_size, void* d_ws, size_t ws_size,
                              hipStream_t stream) {
  (void)in_sizes; (void)n_in; (void)out_size; (void)ws_size;

  float* ws   = (float*)d_ws;
  float* fwd0 = ws;
  float* inv0 = fwd0 + MP0 * T0;
  float* fwd1 = inv0 + (size_t)T0 * KP0;
  float* inv1 = fwd1 + MP1 * T1;
  float* fwd2 = inv1 + (size_t)T1 * KP1;
  float* inv2 = fwd2 + MP2 * T2;

  gen_fwd_basis<<<(MP0*T0 + 255)/256, 256, 0, stream>>>(fwd0, FR0, MP0, T0);
  gen_inv_basis<<<(T0*KP0 + 255)/256, 256, 0, stream>>>(inv0, FR0, KP0, T0);
  gen_fwd_basis<<<(MP1*T1 + 255)/256, 256, 0, stream>>>(fwd1, FR1, MP1, T1);
  gen_inv_basis<<<(T1*KP1 + 255)/256, 256, 0, stream>>>(inv1, FR1, KP1, T1);
  gen_fwd_basis<<<(MP2*T2 + 255)/256, 256, 0, stream>>>(fwd2, FR2, MP2, T2);
  gen_inv_basis<<<(T2*KP2 + 255)/256, 256, 0, stream>>>(inv2, FR2, KP2, T2);

  Params p;
  p.x     = (const float*)d_in[0];
  p.w_emb = (const float*)d_in[1];
  p.b_emb = (const float*)d_in[2];
  int idx = 3;
  for (int l = 0; l < 2; ++l)
    for (int i = 0; i < 2; ++i) {
      p.fw1[l][i] = (const float*)d_in[idx++];
      p.fb1[l][i] = (const float*)d_in[idx++];
      p.fw2[l][i] = (const float*)d_in[idx++];
      p.fb2[l][i] = (const float*)d_in[idx++];
    }
  for (int s = 0; s < 3; ++s) {
    p.c1w[s] = (const float*)d_in[idx++];
    p.c1b[s] = (const float*)d_in[idx++];
    p.c2w[s] = (const float*)d_in[idx++];
    p.c2b[s] = (const float*)d_in[idx++];
  }
  p.fwdB[0] = fwd0; p.fwdB[1] = fwd1; p.fwdB[2] = fwd2;
  p.invB[0] = inv0; p.invB[1] = inv1; p.invB[2] = inv2;
  p.out = (float*)d_out;

  const int smem_bytes = (ROWS_E*LSTR*2 + ROWS_SPEC*LSTR + 64 + 16) * (int)sizeof(float);
  (void)hipFuncSetAttribute((const void*)timemixer_kernel,
                            hipFuncAttributeMaxDynamicSharedMemorySize, smem_bytes);
  timemixer_kernel<<<dim3(BB * NN), dim3(256), smem_bytes, stream>>>(p);
}